// STGNN_65068754534512
// MI455X (gfx1250) — compile-verified
//
#include <hip/hip_runtime.h>
#include <hip/hip_bf16.h>

typedef __attribute__((ext_vector_type(2))) float v2f;
typedef __attribute__((ext_vector_type(8))) float v8f;

#define EPS 1e-5f

// Output modes for the fused GEMM epilogue
#define OUT_PLAIN 0   // C = A@B
#define OUT_BIAS  1   // C = A@B + bias[n]
#define OUT_GCN   2   // C = A@B ; C2 = dinv[row]^2 * C + bias[n]   (self-loop init)

// ---------------------------------------------------------------------------
// fp32 WMMA GEMM, compile-time K, N = NTILES*16, optional fused BN+ReLU on the
// A operand (per-channel scale/shift, channel == K index) and fused GCN
// self-loop epilogue. One wave owns a 16-row strip across the full N extent:
// the A fragment is loaded once per K-step (one unconditional v2f load, plus
// 2 fma+max when BN_A) and reused for all NTILES v_wmma_f32_16x16x4_f32.
// No lane-divergent guards -> EXEC all-ones at every WMMA (ISA requirement).
// ---------------------------------------------------------------------------
template <int K, int NTILES, bool BN_A, int MODE>
__global__ void __launch_bounds__(256)
gemm_wmma(const float* __restrict__ A, const float* __restrict__ Bm,
          const float* __restrict__ bnp,   // [64]: scale[0:32], shift[32:64] (BN_A)
          const float* __restrict__ bias,  // [N]  (OUT_BIAS / OUT_GCN)
          const float* __restrict__ dinv,  // [M]  (OUT_GCN)
          float* __restrict__ C, float* __restrict__ C2,
          int M) {
    constexpr int N = NTILES * 16;
    const int lane = threadIdx.x & 31;
    const int wave = blockIdx.x * (blockDim.x >> 5) + (threadIdx.x >> 5);
    if (wave * 16 >= M) return;                 // wave-uniform exit

    const int l16   = lane & 15;
    const int hi    = lane >> 4;                // 0 or 1
    const int khalf = hi * 2;                   // A/B fragment K-offset
    const int arow  = wave * 16 + l16;

    v8f c[NTILES];
#pragma unroll
    for (int i = 0; i < NTILES; ++i) c[i] = (v8f){0.f,0.f,0.f,0.f,0.f,0.f,0.f,0.f};

    const float* __restrict__ Arow = A + (size_t)arow * K;

#pragma unroll
    for (int k0 = 0; k0 < K; k0 += 4) {
        const int ka = k0 + khalf;
        // 8-byte aligned pair load: khalf even, k0 multiple of 4
        v2f a = *(const v2f*)(Arow + ka);
        if (BN_A) {                              // fused BatchNorm + ReLU on A
            a.x = fmaxf(a.x * bnp[ka]     + bnp[32 + ka],     0.f);
            a.y = fmaxf(a.y * bnp[ka + 1] + bnp[32 + ka + 1], 0.f);
        }
#pragma unroll
        for (int nt = 0; nt < NTILES; ++nt) {
            const int n = nt * 16 + l16;
            v2f b;
            b.x = Bm[ka * N + n];
            b.y = Bm[(ka + 1) * N + n];
            c[nt] = __builtin_amdgcn_wmma_f32_16x16x4_f32(
                        false, a, false, b, (short)0, c[nt], false, false);
        }
    }

    const int rowbase = wave * 16 + hi * 8;     // D layout: vgpr r -> row r + 8*hi
    float sw[8];
    if (MODE == OUT_GCN) {
#pragma unroll
        for (int r = 0; r < 8; ++r) {
            float dv = dinv[rowbase + r];
            sw[r] = dv * dv;
        }
    }
#pragma unroll
    for (int nt = 0; nt < NTILES; ++nt) {
        const int n = nt * 16 + l16;
        const float bv = (MODE != OUT_PLAIN) ? bias[n] : 0.f;
#pragma unroll
        for (int r = 0; r < 8; ++r) {
            const size_t off = (size_t)(rowbase + r) * N + n;
            const float cv = c[nt][r];
            if (MODE == OUT_GCN) {
                C[off]  = cv;                    // raw product for edge scatter
                C2[off] = sw[r] * cv + bv;       // self-loop + bias seed for agg
            } else if (MODE == OUT_BIAS) {
                C[off] = cv + bv;
            } else {
                C[off] = cv;
            }
        }
    }
}

// ---------------------------------------------------------------------------
// Helper kernels
// ---------------------------------------------------------------------------
__global__ void zero_f32(float* p, int n) {
    int i = blockIdx.x * blockDim.x + threadIdx.x;
    if (i < n) p[i] = 0.f;
}

__global__ void deg_count(const int* __restrict__ dst, float* __restrict__ deg, int E) {
    int e = blockIdx.x * blockDim.x + threadIdx.x;
    if (e < E) atomicAdd(&deg[dst[e]], 1.f);
}

__global__ void finalize_dinv(const float* __restrict__ deg, float* __restrict__ dinv, int BN) {
    int i = blockIdx.x * blockDim.x + threadIdx.x;
    if (i < BN) dinv[i] = rsqrtf(deg[i] + 1.f);
}

// W[R,Ccols] -> WT[Ccols,R]
__global__ void transpose_w(const float* __restrict__ W, float* __restrict__ WT,
                            int R, int Ccols) {
    int idx = blockIdx.x * blockDim.x + threadIdx.x;
    if (idx >= R * Ccols) return;
    int r = idx / Ccols, c = idx - r * Ccols;
    WT[c * R + r] = W[idx];
}

// Pad W1 [7,32] -> W1p [8,32] (zero last row)
__global__ void pad_w1(const float* __restrict__ W1, float* __restrict__ W1p) {
    int idx = blockIdx.x * blockDim.x + threadIdx.x;
    if (idx >= 8 * 32) return;
    int r = idx >> 5;
    W1p[idx] = (r < 7) ? W1[idx] : 0.f;
}

// xtp[m*8+f] = x[b, t, n, f] for f<7, 0 for f==7  (m = b*N + n)
__global__ void slice_xt_pad(const float* __restrict__ x, float* __restrict__ xt,
                             int t, int B, int T, int N, int Fin) {
    int idx = blockIdx.x * blockDim.x + threadIdx.x;
    if (idx >= B * N * 8) return;
    int f = idx & 7;
    int m = idx >> 3;
    int b = m / N, n = m - b * N;
    xt[idx] = (f < Fin) ? x[(((size_t)b * T + t) * N + n) * Fin + f] : 0.f;
}

// out[dst] += tmp[src] * (dinv[src]*dinv[dst]); one thread = one edge x 8 channels
__global__ void edge_agg(const int* __restrict__ src, const int* __restrict__ dst,
                         const float* __restrict__ dinv, const float* __restrict__ tmp,
                         float* __restrict__ out, int E, int H) {
    int gid = blockIdx.x * blockDim.x + threadIdx.x;
    if (gid >= E * 4) return;
    int e  = gid >> 2;
    int j0 = (gid & 3) * 8;
    int s = src[e], d = dst[e];
    float w = dinv[s] * dinv[d];
    const float* ts = tmp + (size_t)s * H + j0;
    float*       od = out + (size_t)d * H + j0;
#pragma unroll
    for (int j = 0; j < 8; ++j) atomicAdd(od + j, ts[j] * w);
}

// per-channel sum / sum-of-squares -> sums[0:32], sums[32:64]
// Each lane privately accumulates its channel over strided rows (coalesced
// 128B wave reads), then two LDS atomics per thread + one global atomic per sums slot per block.
__global__ void bn_reduce(const float* __restrict__ h, float* __restrict__ sums, int BN) {
    __shared__ float ls[64];
    if (threadIdx.x < 64) ls[threadIdx.x] = 0.f;
    __syncthreads();
    const int j = threadIdx.x & 31;
    int row = (blockIdx.x * blockDim.x + threadIdx.x) >> 5;
    const int rstride = (gridDim.x * blockDim.x) >> 5;
    float s = 0.f, sq = 0.f;
    for (int r = row; r < BN; r += rstride) {
        float v = h[(size_t)r * 32 + j];
        s += v; sq += v * v;
    }
    atomicAdd(&ls[j], s);
    atomicAdd(&ls[32 + j], sq);
    __syncthreads();
    if (threadIdx.x < 64) atomicAdd(&sums[threadIdx.x], ls[threadIdx.x]);
}

// sums -> per-channel affine: bnp[j] = g*rsqrt(var+eps), bnp[32+j] = be - mean*scale
__global__ void bn_finalize(const float* __restrict__ sums, float* __restrict__ bnp,
                            const float* __restrict__ g, const float* __restrict__ be,
                            int BN) {
    int j = threadIdx.x;
    if (j >= 32) return;
    float mean = sums[j] / (float)BN;
    float var  = sums[32 + j] / (float)BN - mean * mean;
    float sc   = g[j] * rsqrtf(var + EPS);
    bnp[j]      = sc;
    bnp[32 + j] = be[j] - mean * sc;
}

// GRU elementwise update, h updated in place
__global__ void gru_elem(const float* __restrict__ gi, const float* __restrict__ gh,
                         float* __restrict__ h, int BN, int H) {
    int idx = blockIdx.x * blockDim.x + threadIdx.x;
    if (idx >= BN * H) return;
    int m = idx / H, j = idx - m * H;
    const float* gim = gi + (size_t)m * 3 * H;
    const float* ghm = gh + (size_t)m * 3 * H;
    float ir  = gim[j],         iz = gim[H + j],  inn = gim[2 * H + j];
    float hr  = ghm[j],         hz = ghm[H + j],  hn  = ghm[2 * H + j];
    float r = 1.f / (1.f + __expf(-(ir + hr)));
    float z = 1.f / (1.f + __expf(-(iz + hz)));
    float nn = tanhf(inn + r * hn);
    float hp = h[idx];
    h[idx] = (1.f - z) * nn + z * hp;
}

// preds = h @ W_fc^T + b_fc   (O=3, tiny -> plain VALU kernel)
__global__ void fc_out(const float* __restrict__ h, const float* __restrict__ Wfc,
                       const float* __restrict__ bfc, float* __restrict__ out,
                       int BN, int H, int O) {
    int idx = blockIdx.x * blockDim.x + threadIdx.x;
    if (idx >= BN * O) return;
    int m = idx / O, o = idx - m * O;
    float acc = bfc[o];
    const float* hm = h + (size_t)m * H;
    const float* wo = Wfc + (size_t)o * H;
#pragma unroll 8
    for (int j = 0; j < H; ++j) acc += hm[j] * wo[j];
    out[idx] = acc;
}

__global__ void mask_kernel(const float* __restrict__ x, float* __restrict__ mask,
                            int total, int Fin) {
    int idx = blockIdx.x * blockDim.x + threadIdx.x;
    if (idx >= total) return;
    const float* xp = x + (size_t)idx * Fin;
    int any = 0;
#pragma unroll
    for (int f = 0; f < 7; ++f) any |= (xp[f] != 0.f);
    mask[idx] = any ? 1.f : 0.f;
}

// ---------------------------------------------------------------------------
// Launch
// ---------------------------------------------------------------------------
static inline int cdiv(int a, int b) { return (a + b - 1) / b; }

extern "C" void kernel_launch(void* const* d_in, const int* in_sizes, int n_in,
                              void* d_out, int out_size, void* d_ws, size_t ws_size,
                              hipStream_t stream) {
    const float* x    = (const float*)d_in[0];
    const int*   ei   = (const int*)  d_in[1];
    const float* W1   = (const float*)d_in[4];
    const float* b1   = (const float*)d_in[5];
    const float* g1   = (const float*)d_in[6];
    const float* be1  = (const float*)d_in[7];
    const float* W2   = (const float*)d_in[8];
    const float* b2   = (const float*)d_in[9];
    const float* g2   = (const float*)d_in[10];
    const float* be2  = (const float*)d_in[11];
    const float* W_ih = (const float*)d_in[12];
    const float* W_hh = (const float*)d_in[13];
    const float* b_ih = (const float*)d_in[14];
    const float* b_hh = (const float*)d_in[15];
    const float* W_fc = (const float*)d_in[16];
    const float* b_fc = (const float*)d_in[17];

    const int B = 16, T = 20, N = 5000, Fin = 7, H = 32, O = 3;
    const int BN = B * N;
    const int E  = in_sizes[1] / 2;
    const int* src = ei;
    const int* dst = ei + E;

    // workspace carve-up (floats)
    float* p = (float*)d_ws;
    float* deg  = p;  p += BN;
    float* dinv = p;  p += BN;
    float* xtp  = p;  p += (size_t)BN * 8;       // padded K=8 input slice
    float* tmp  = p;  p += (size_t)BN * H;       // raw GEMM product (edge scatter src)
    float* hcur = p;  p += (size_t)BN * H;       // layer-1 aggregation buffer
    float* h2   = p;  p += (size_t)BN * H;       // layer-2 aggregation buffer
    float* gi   = p;  p += (size_t)BN * 3 * H;
    float* gh   = p;  p += (size_t)BN * 3 * H;
    float* hgru = p;  p += (size_t)BN * H;
    float* sums = p;  p += 2 * H;
    float* bnp1 = p;  p += 2 * H;                // BN scale/shift layer 1
    float* bnp2 = p;  p += 2 * H;                // BN scale/shift layer 2
    float* W1p  = p;  p += 8 * H;                // padded [8,32]
    float* WihT = p;  p += 3 * H * H;
    float* WhhT = p;  p += 3 * H * H;

    const int TB = 256;

    // ---- degree / dinv ----
    zero_f32<<<cdiv(BN, TB), TB, 0, stream>>>(deg, BN);
    deg_count<<<cdiv(E, TB), TB, 0, stream>>>(dst, deg, E);
    finalize_dinv<<<cdiv(BN, TB), TB, 0, stream>>>(deg, dinv, BN);

    // ---- weight prep ----
    pad_w1<<<1, 256, 0, stream>>>(W1, W1p);
    transpose_w<<<cdiv(3 * H * H, TB), TB, 0, stream>>>(W_ih, WihT, 3 * H, H);
    transpose_w<<<cdiv(3 * H * H, TB), TB, 0, stream>>>(W_hh, WhhT, 3 * H, H);

    // ---- GRU hidden init ----
    zero_f32<<<cdiv(BN * H, TB), TB, 0, stream>>>(hgru, BN * H);

    // GEMM grids: one wave per 16-row strip; 8 waves per 256-thread block
    const int gemm_blocks = cdiv(BN / 16, 8);    // 625 blocks
    const int nh_blocks   = cdiv(BN * H, TB);
    const int e4_blocks   = cdiv(E * 4, TB);
    const int red_blocks  = 512;

    for (int t = 0; t < T; ++t) {
        slice_xt_pad<<<cdiv(BN * 8, TB), TB, 0, stream>>>(x, xtp, t, B, T, N, Fin);

        // ---- GCN layer 1: tmp = xtp@W1p ; hcur = sw*tmp + b1 (fused) ----
        gemm_wmma<8, 2, false, OUT_GCN><<<gemm_blocks, TB, 0, stream>>>(
            xtp, W1p, nullptr, b1, dinv, tmp, hcur, BN);
        edge_agg<<<e4_blocks, TB, 0, stream>>>(src, dst, dinv, tmp, hcur, E, H);
        zero_f32<<<1, 64, 0, stream>>>(sums, 2 * H);
        bn_reduce<<<red_blocks, TB, 0, stream>>>(hcur, sums, BN);
        bn_finalize<<<1, 32, 0, stream>>>(sums, bnp1, g1, be1, BN);

        // ---- GCN layer 2: tmp = relu(bn1(hcur))@W2 ; h2 = sw*tmp + b2 (fused) ----
        gemm_wmma<32, 2, true, OUT_GCN><<<gemm_blocks, TB, 0, stream>>>(
            hcur, W2, bnp1, b2, dinv, tmp, h2, BN);
        edge_agg<<<e4_blocks, TB, 0, stream>>>(src, dst, dinv, tmp, h2, E, H);
        zero_f32<<<1, 64, 0, stream>>>(sums, 2 * H);
        bn_reduce<<<red_blocks, TB, 0, stream>>>(h2, sums, BN);
        bn_finalize<<<1, 32, 0, stream>>>(sums, bnp2, g2, be2, BN);

        // ---- GRU step: gi = relu(bn2(h2))@WihT + b_ih ; gh = hgru@WhhT + b_hh ----
        gemm_wmma<32, 6, true, OUT_BIAS><<<gemm_blocks, TB, 0, stream>>>(
            h2, WihT, bnp2, b_ih, nullptr, gi, nullptr, BN);
        gemm_wmma<32, 6, false, OUT_BIAS><<<gemm_blocks, TB, 0, stream>>>(
            hgru, WhhT, nullptr, b_hh, nullptr, gh, nullptr, BN);
        gru_elem<<<nh_blocks, TB, 0, stream>>>(gi, gh, hgru, BN, H);
    }

    // ---- outputs: preds [BN,3] then mask [B,T,N] ----
    float* out = (float*)d_out;
    fc_out<<<cdiv(BN * O, TB), TB, 0, stream>>>(hgru, W_fc, b_fc, out, BN, H, O);
    mask_kernel<<<cdiv(B * T * N, TB), TB, 0, stream>>>(x, out + (size_t)BN * O, B * T * N, Fin);
}